// HeteroGNN_17721035063558
// MI455X (gfx1250) — compile-verified
//
#include <hip/hip_runtime.h>
#include <hip/hip_bf16.h>

typedef __attribute__((ext_vector_type(2))) float v2f;
typedef __attribute__((ext_vector_type(8))) float v8f;

#define N_NODES 100000
#define HIDC    128

// ---------------------------------------------------------------- zero fill
__global__ void sage_zero_f4(float4* p, int n4) {
    int i = blockIdx.x * blockDim.x + threadIdx.x;
    if (i < n4) p[i] = make_float4(0.f, 0.f, 0.f, 0.f);
}

// ---------------------------------------------------------------- degree count
__global__ void sage_count(const int* __restrict__ ei, float* __restrict__ cnt, int E) {
    int e = blockIdx.x * blockDim.x + threadIdx.x;
    if (e < E) {
        int d = ei[E + e];                 // dst row of edge_index
        atomicAdd(&cnt[d], 1.0f);
    }
}

// ---------------------------------------------------------------- scatter-add (one wave per edge)
// F = 64: float2 per lane; F = 128: float4 per lane. Fully coalesced row read,
// contiguous f32 atomic adds on the destination row.
template <int F>
__global__ void sage_scatter(const float* __restrict__ feat,
                             const int* __restrict__ ei,
                             float* __restrict__ agg, int E) {
    const int lane = threadIdx.x & 31;
    const int wave = threadIdx.x >> 5;
    long long e = (long long)blockIdx.x * (blockDim.x >> 5) + wave;
    if (e >= E) return;
    int s = ei[e];
    int d = ei[E + e];
    constexpr int PER = F / 32;            // floats per lane
    const float* fp = feat + (long long)s * F + lane * PER;
    float*       ap = agg  + (long long)d * F + lane * PER;
    float v[PER];
#pragma unroll
    for (int i = 0; i < PER; ++i) v[i] = fp[i];
#pragma unroll
    for (int i = 0; i < PER; ++i) atomicAdd(&ap[i], v[i]);
}

// ---------------------------------------------------------------- fused SAGE node transform
// out[16 nodes x 128] = (agg/max(cnt,1)) @ Wl^T + x_self @ Wr^T + bl  [, ReLU]
// Block: 128 threads (4 waves). Wave w computes column tiles {w, w+4}.
// WMMA: V_WMMA_F32_16X16X4_F32, K-loop in steps of 4.
template <int K, bool RELU>
__global__ void sage_node_gemm(const float* __restrict__ agg,
                               const float* __restrict__ cnt,
                               const float* __restrict__ xself,
                               const float* __restrict__ Wl,   // [128, K]
                               const float* __restrict__ Wr,   // [128, K]
                               const float* __restrict__ bl,   // [128]
                               float* __restrict__ out) {      // [N, 128]
    constexpr int LDW = K + 2;             // even stride: 8B-aligned b64 LDS loads,
                                           // banks = 2m for lanes 0..15 -> conflict free
    __shared__ float sMean[16 * LDW];
    __shared__ float sSelf[16 * LDW];

    const int nodeBase = blockIdx.x * 16;

    // cooperative stage: mean + self features into LDS (coalesced global reads)
    for (int idx = threadIdx.x; idx < 16 * K; idx += blockDim.x) {
        int r = idx / K;                   // K is a power of two -> shift
        int c = idx - r * K;
        int node = nodeBase + r;
        float inv = 1.0f / fmaxf(cnt[node], 1.0f);
        sMean[r * LDW + c] = agg[(long long)node * K + c] * inv;
        sSelf[r * LDW + c] = xself[(long long)node * K + c];
    }
    __syncthreads();

    const int lane = threadIdx.x & 31;
    const int wv   = threadIdx.x >> 5;     // 0..3
    const int mn   = lane & 15;            // row m (A) / col n (B,C,D)
    const int h    = lane >> 4;            // half-wave select

#pragma unroll
    for (int t = 0; t < 2; ++t) {
        const int nt  = wv + 4 * t;        // column tile 0..7
        const int col = nt * 16 + mn;
        const float bias = bl[col];
        v8f acc = {bias, bias, bias, bias, bias, bias, bias, bias};

        const float* wlrow = Wl + (long long)col * K;   // B[k][n] = W[col][k]
        const float* wrrow = Wr + (long long)col * K;

#pragma unroll
        for (int k0 = 0; k0 < K; k0 += 4) {
            const int kk = k0 + 2 * h;     // lane's K pair per ISA A/B layout
            v2f a1 = *(const v2f*)&sMean[mn * LDW + kk];
            v2f b1 = *(const v2f*)&wlrow[kk];
            acc = __builtin_amdgcn_wmma_f32_16x16x4_f32(
                false, a1, false, b1, (short)0, acc, false, false);
            v2f a2 = *(const v2f*)&sSelf[mn * LDW + kk];
            v2f b2 = *(const v2f*)&wrrow[kk];
            acc = __builtin_amdgcn_wmma_f32_16x16x4_f32(
                false, a2, false, b2, (short)0, acc, false, false);
        }

        // C/D layout: VGPR r -> M = r + 8*h, N = lane%16
#pragma unroll
        for (int r = 0; r < 8; ++r) {
            int row = nodeBase + r + 8 * h;
            float v = acc[r];
            if (RELU) v = fmaxf(v, 0.0f);
            out[(long long)row * HIDC + col] = v;
        }
    }
}

// ---------------------------------------------------------------- launcher
extern "C" void kernel_launch(void* const* d_in, const int* in_sizes, int n_in,
                              void* d_out, int out_size, void* d_ws, size_t ws_size,
                              hipStream_t stream) {
    const float* x    = (const float*)d_in[0];   // [N, 64]
    const int*   ei   = (const int*)d_in[1];     // [2, E] flat
    const float* Wl1  = (const float*)d_in[2];   // [128, 64]
    const float* bl1  = (const float*)d_in[3];   // [128]
    const float* Wr1  = (const float*)d_in[4];   // [128, 64]
    const float* Wl2  = (const float*)d_in[5];   // [128, 128]
    const float* bl2  = (const float*)d_in[6];   // [128]
    const float* Wr2  = (const float*)d_in[7];   // [128, 128]
    float* out = (float*)d_out;

    const int E = in_sizes[1] / 2;               // 1,600,000

    // workspace layout (floats): cnt | h | agg (agg reused for both layers)
    float* cnt = (float*)d_ws;                   // 100096 (padded)
    float* h   = cnt + 100096;                   // N*128
    float* agg = h + (size_t)N_NODES * HIDC;     // N*128

    const int nAggF4 = (N_NODES * HIDC) / 4;     // 3,200,000
    const int nCntF4 = 100096 / 4;

    // 1) zero counts + aggregate buffer
    sage_zero_f4<<<(nCntF4 + 255) / 256, 256, 0, stream>>>((float4*)cnt, nCntF4);
    sage_zero_f4<<<(nAggF4 + 255) / 256, 256, 0, stream>>>((float4*)agg, nAggF4);

    // 2) degree counts
    sage_count<<<(E + 255) / 256, 256, 0, stream>>>(ei, cnt, E);

    // 3) layer-1 scatter: agg[dst] += x[src]   (F=64, wave per edge, 8 waves/block)
    sage_scatter<64><<<(E + 7) / 8, 256, 0, stream>>>(x, ei, agg, E);

    // 4) layer-1 fused mean-GEMM + bias + ReLU -> h
    sage_node_gemm<64, true><<<N_NODES / 16, 128, 0, stream>>>(
        agg, cnt, x, Wl1, Wr1, bl1, h);

    // 5) re-zero aggregate buffer, layer-2 scatter: agg[dst] += h[src] (F=128)
    sage_zero_f4<<<(nAggF4 + 255) / 256, 256, 0, stream>>>((float4*)agg, nAggF4);
    sage_scatter<128><<<(E + 7) / 8, 256, 0, stream>>>(h, ei, agg, E);

    // 6) layer-2 fused mean-GEMM + bias -> out
    sage_node_gemm<128, false><<<N_NODES / 16, 128, 0, stream>>>(
        agg, cnt, h, Wl2, Wr2, bl2, out);

    (void)n_in; (void)out_size; (void)ws_size;
}